// CTCLoss_67216238182819
// MI455X (gfx1250) — compile-verified
//
#include <hip/hip_runtime.h>
#include <hip/hip_bf16.h>

// ---------------- problem constants (match reference setup_inputs) ----------
#define BATCH 32
#define TLEN  1600
#define CDIM  1024
#define LLEN  128
#define SDIM  257          // 2*L + 1
#define SPAD  264          // padded LDS row for alpha
#define NEGV  (-1e30f)
#define LOG2E 1.4426950408889634f
#define LN2   0.6931471805599453f

typedef __attribute__((ext_vector_type(2))) float v2f;
typedef __attribute__((ext_vector_type(8))) float v8f;

// Raw-hardware transcendentals (v_exp_f32 / v_log_f32, no OCML denorm fixup).
// Safe here: exp2 args are <= 0 (flush-to-zero of huge negatives is exactly
// what log-space "zero" needs) and log2 args are >= 1 by construction.
__device__ __forceinline__ float fexp2(float x) { return __builtin_amdgcn_exp2f(x); }
__device__ __forceinline__ float flog2(float x) { return __builtin_amdgcn_logf(x); }

// ---------------------------------------------------------------------------
// Kernel 1: per-row logsumexp over C=1024.  One block per (b,t) row.
// Each thread loads one float4 (coalesced b128), computes local max, block max
// via wave shuffles + LDS, then sum of exp2.  The 32-lane partial-sum
// reduction is done with V_WMMA_F32_16X16X4_F32 against a ones matrix
// (D[i][j] = p[i] + p[i+16]); 8 accumulator VGPR adds + one xor-16 shuffle
// complete the wave sum.  Deterministic (fixed reduction order, no atomics).
// ---------------------------------------------------------------------------
__global__ __launch_bounds__(256) void lse_rows_kernel(
    const float* __restrict__ lp, float* __restrict__ lse) {
  const int row = blockIdx.x;                 // row = b*T + t
  const int tid = threadIdx.x;
  const int wid = tid >> 5, lane = tid & 31;

  const float4 x = ((const float4*)(lp + (size_t)row * CDIM))[tid];

  // ---- block max ----
  float m = fmaxf(fmaxf(x.x, x.y), fmaxf(x.z, x.w));
#pragma unroll
  for (int off = 16; off > 0; off >>= 1)
    m = fmaxf(m, __shfl_xor(m, off, 32));

  __shared__ float smax[8];
  __shared__ float ssum[8];
  if (lane == 0) smax[wid] = m;
  __syncthreads();
  float M = smax[0];
#pragma unroll
  for (int w = 1; w < 8; ++w) M = fmaxf(M, smax[w]);

  // ---- per-lane partial sum of exp ----
  float p = fexp2((x.x - M) * LOG2E) + fexp2((x.y - M) * LOG2E) +
            fexp2((x.z - M) * LOG2E) + fexp2((x.w - M) * LOG2E);

  // ---- wave32 sum via WMMA f32 16x16x4 against ones ----
  v2f a;  a.x = p;    a.y = 0.0f;      // A[i][0]=p[i], A[i][2]=p[i+16]
  v2f bo; bo.x = 1.0f; bo.y = 1.0f;    // B = all ones
  v8f acc = {};
  acc = __builtin_amdgcn_wmma_f32_16x16x4_f32(
      /*neg_a=*/false, a, /*neg_b=*/false, bo,
      /*c_mod=*/(short)0, acc, /*reuse_a=*/false, /*reuse_b=*/false);
  float d = acc[0] + acc[1] + acc[2] + acc[3] +
            acc[4] + acc[5] + acc[6] + acc[7];   // half-wave row sums
  d += __shfl_xor(d, 16, 32);                    // full wave sum, all lanes

  if (lane == 0) ssum[wid] = d;
  __syncthreads();
  if (tid == 0) {
    float s = 0.0f;
#pragma unroll
    for (int w = 0; w < 8; ++w) s += ssum[w];
    lse[row] = M + flog2(s) * LN2;
  }
}

// ---------------------------------------------------------------------------
// Kernel 2: norm[b] = sum_{t < input_len[b]} lse[b,t]  (deterministic tree)
// ---------------------------------------------------------------------------
__global__ __launch_bounds__(256) void norm_kernel(
    const float* __restrict__ lse, const int* __restrict__ input_lengths,
    float* __restrict__ norm) {
  const int b = blockIdx.x;
  const int Tin = input_lengths[b];
  float acc = 0.0f;
  for (int t = threadIdx.x; t < Tin; t += 256) acc += lse[b * TLEN + t];
  __shared__ float red[256];
  red[threadIdx.x] = acc;
  __syncthreads();
#pragma unroll
  for (int off = 128; off > 0; off >>= 1) {
    if (threadIdx.x < off) red[threadIdx.x] += red[threadIdx.x + off];
    __syncthreads();
  }
  if (threadIdx.x == 0) norm[b] = red[0];
}

// ---------------------------------------------------------------------------
// Kernel 3: CTC forward recursion.  One block per batch element, thread s owns
// extended state s (s < 257; 288 threads = 9 full waves).  alpha double-
// buffered in LDS -> exactly one __syncthreads per time step.  Emission
// gathers (fixed per-thread column ext[s], stride C) are software-pipelined
// 8 steps deep in a register ring.  Loop stops at input_length[b] (alpha is
// frozen past it in the reference).
// ---------------------------------------------------------------------------
__global__ __launch_bounds__(288) void ctc_forward_kernel(
    const float* __restrict__ lp, const int* __restrict__ targets,
    const int* __restrict__ input_lengths, const int* __restrict__ target_lengths,
    const float* __restrict__ norm, float* __restrict__ out) {
  const int b = blockIdx.x;
  const int s = threadIdx.x;
  const int Tin = input_lengths[b];
  const int Sb  = 2 * target_lengths[b] + 1;

  __shared__ float A[2][SPAD];

  const bool active = (s < SDIM);
  bool  skip = false, valid = false;
  const float* lpb = nullptr;

  if (active) {
    valid = (s < Sb);
    int ext = 0;
    if (s & 1) {
      const int li = s >> 1;
      ext = targets[b * LLEN + li];
      if (s >= 3) skip = (ext != targets[b * LLEN + li - 1]);
    }
    lpb = lp + (size_t)b * TLEN * CDIM + ext;
  }

  // t = 0 init: only states 0,1 reachable
  float a0 = NEGV;
  if (active) {
    a0 = (s < 2 && valid) ? lpb[0] : NEGV;
    A[0][s] = a0;
  }

  // prime the 8-deep emission prefetch ring (emissions for t = 1..8)
  float ebuf[8];
  if (active) {
#pragma unroll
    for (int k = 0; k < 8; ++k)
      ebuf[k] = (1 + k < Tin) ? lpb[(size_t)(1 + k) * CDIM] : NEGV;
  }
  __syncthreads();

  int cur = 0;
  for (int t = 1; t < Tin; ++t) {
    float na = NEGV;
    if (active) {
      const float e = ebuf[(t - 1) & 7];
      const int tpf = t + 8;
      if (tpf < Tin) ebuf[(t - 1) & 7] = lpb[(size_t)tpf * CDIM];

      const float a1 = (s >= 1) ? A[cur][s - 1] : NEGV;
      const float a2 = skip ? A[cur][s - 2] : NEGV;
      const float m  = fmaxf(a0, fmaxf(a1, a2));
      const float su = fexp2((a0 - m) * LOG2E) + fexp2((a1 - m) * LOG2E) +
                       fexp2((a2 - m) * LOG2E);
      na = valid ? (e + m + flog2(su) * LN2) : NEGV;
      A[cur ^ 1][s] = na;
    }
    __syncthreads();
    a0 = na;
    cur ^= 1;
  }

  if (s == 0) {
    const float x = A[cur][Sb - 1];
    const float y = A[cur][Sb - 2];
    const float m = fmaxf(x, y);
    const float llh =
        m + flog2(fexp2((x - m) * LOG2E) + fexp2((y - m) * LOG2E)) * LN2;
    out[b] = norm[b] - llh;
  }
}

// ---------------------------------------------------------------------------
extern "C" void kernel_launch(void* const* d_in, const int* in_sizes, int n_in,
                              void* d_out, int out_size, void* d_ws, size_t ws_size,
                              hipStream_t stream) {
  const float* lp       = (const float*)d_in[0];   // [B,T,C] f32 (log-softmaxed)
  const int*   targets  = (const int*)d_in[1];     // [B,L]
  const int*   in_len   = (const int*)d_in[2];     // [B]
  const int*   tg_len   = (const int*)d_in[3];     // [B]
  float*       out      = (float*)d_out;           // [B] f32

  float* ws_lse  = (float*)d_ws;                   // B*T floats
  float* ws_norm = ws_lse + BATCH * TLEN;          // B floats

  lse_rows_kernel<<<BATCH * TLEN, 256, 0, stream>>>(lp, ws_lse);
  norm_kernel<<<BATCH, 256, 0, stream>>>(ws_lse, in_len, ws_norm);
  ctc_forward_kernel<<<BATCH, 288, 0, stream>>>(lp, targets, in_len, tg_len,
                                                ws_norm, out);
}